// RAFilterTorch_30683246362699
// MI455X (gfx1250) — compile-verified
//
#include <hip/hip_runtime.h>
#include <math.h>

// Problem shape (fixed by reference setup_inputs)
#define B_   8
#define T_   4096
#define N_   2048
#define BN   (B_ * N_)          // 16384 independent (b,n) sequences
#define CCH  16                 // time chunks
#define LCH  (T_ / CCH)         // 256 timesteps per chunk
#define NB   (LCH / 16)         // 16-step WMMA blocks per chunk

#define A_DEC 0.99666666666666667f   // 1 - dt/tau
#define KD    0.06666666666666667f   // K3/tau

typedef __attribute__((ext_vector_type(2))) float v2f;
typedef __attribute__((ext_vector_type(4))) float v4f;
typedef __attribute__((ext_vector_type(8))) float v8f;

// ---------------------------------------------------------------------------
// Pass 1: per-(column, chunk) final state of the zero-initialized local scan.
// 4 columns per thread via b128 loads: 512B/wave-instruction, 2048 waves.
// ---------------------------------------------------------------------------
__global__ void __launch_bounds__(256)
ra_finals_kernel(const float* __restrict__ in, float* __restrict__ fin) {
    const int col4  = blockIdx.x * 256 + threadIdx.x;    // 0..BN/4-1
    const int chunk = blockIdx.y;                         // 0..CCH-1
    const int col   = col4 * 4;
    const int b = col / N_;
    const int n = col - b * N_;
    const size_t base = (size_t)b * T_ * N_ + n;
    const int t0 = chunk * LCH;
    const float* p = in + base + (size_t)t0 * N_;

    v4f prev = {0.f, 0.f, 0.f, 0.f};
    if (t0 > 0) prev = *(const v4f*)(p - N_);             // I[-1] == 0
    v4f s = {0.f, 0.f, 0.f, 0.f};
#pragma unroll 8
    for (int t = 0; t < LCH; ++t) {
        v4f x = *(const v4f*)(p + (size_t)t * N_);
#pragma unroll
        for (int j = 0; j < 4; ++j)
            s[j] = fmaf(A_DEC, s[j], KD * fabsf(x[j] - prev[j]));
        prev = x;
    }
    *(v4f*)(fin + (size_t)chunk * BN + col) = s;
}

// ---------------------------------------------------------------------------
// Pass 2: one wave per (16-column tile, chunk). Each 16-step block is
// D = L x U on the WMMA unit (4x v_wmma_f32_16x16x4_f32), L[i][j] = a^(i-j).
// Input staged global->LDS with async b128 transfers, double-buffered so the
// next block's HBM fetch overlaps the current block's WMMA/VALU work.
// All hot-loop addresses are carried incrementally so loads/stores keep one
// base register + immediate offsets (clause-friendly).
// ---------------------------------------------------------------------------
#define WPB 8                   // waves per block

__global__ void __launch_bounds__(32 * WPB)
ra_scan_kernel(const float* __restrict__ in,
               const float* __restrict__ fin,
               float* __restrict__ out) {
    __shared__ float lds[WPB][2][256];                    // double buffer/wave

    const int lane = threadIdx.x & 31;
    const int wv   = threadIdx.x >> 5;
    const int tile = blockIdx.x * WPB + wv;               // 0 .. BN/16*CCH - 1
    const int NUMCT = BN / 16;                            // 1024 column tiles
    const int colTile = tile % NUMCT;
    // Reverse chunk order: high-t chunks launch first and hit the input tail
    // that pass 1 left resident in the 192MB L2.
    const int chunk = (CCH - 1) - tile / NUMCT;

    const int m    = lane & 15;                           // column / A row
    const int half = lane >> 4;                           // lo/hi half-wave

    const int col0 = colTile * 16;
    const int b    = col0 / N_;
    const int n0   = col0 - b * N_;
    const size_t base = (size_t)b * T_ * N_ + n0;
    const int t0 = chunk * LCH;

    // Per-lane staging coordinates: 2 async b128 ops cover 16 rows x 16 cols.
    const int sr = lane >> 2;                             // row 0..7
    const int sc = (lane & 3) * 4;                        // col 0,4,8,12

    // Incremental address state for the hot loop.
    const float* gq0 = in + base + (size_t)(t0 + sr) * N_ + sc;   // rows 0..7
    const float* gq1 = gq0 + (size_t)8 * N_;                      // rows 8..15
    const unsigned laA0 = (unsigned)(size_t)&lds[wv][0][sr * 16 + sc];
    const unsigned laA1 = laA0 + 512;
    const unsigned laB0 = (unsigned)(size_t)&lds[wv][1][sr * 16 + sc];
    const unsigned laB1 = laB0 + 512;
    float* op = out + base + (size_t)(t0 + 8 * half) * N_ + m;    // store base

    // ---- prologue: async-stage block 0 into buffer 0 -----------------------
    asm volatile("global_load_async_to_lds_b128 %0, %1, off"
                 :: "v"(laA0), "v"((unsigned long long)(size_t)gq0) : "memory");
    asm volatile("global_load_async_to_lds_b128 %0, %1, off"
                 :: "v"(laA1), "v"((unsigned long long)(size_t)gq1) : "memory");
    gq0 += (size_t)16 * N_;
    gq1 += (size_t)16 * N_;

    // Boundary row I[t0-1] for the drive at the chunk's first step.
    float prevm = 0.0f;
    if (t0 > 0) prevm = in[base + (size_t)(t0 - 1) * N_ + m];

    // ---- incoming carry from pass-1 finals (Horner in a^L) -----------------
    const float aL = __powf(A_DEC, (float)LCH);
    float s = 0.0f;
    for (int j = 0; j < chunk; ++j)
        s = fmaf(aL, s, fin[(size_t)j * BN + col0 + m]);

    // ---- per-lane A-matrix constants: L[m][c], c = 4*k4 + 2*half (+1) ------
    float Av0[4], Av1[4];
#pragma unroll
    for (int k4 = 0; k4 < 4; ++k4) {
        const int c0 = 4 * k4 + 2 * half;
        const int c1 = c0 + 1;
        float p0 = 0.0f, p1 = 0.0f;
        if (c0 <= m) { p0 = 1.0f; for (int e = 0; e < m - c0; ++e) p0 *= A_DEC; }
        if (c1 <= m) { p1 = 1.0f; for (int e = 0; e < m - c1; ++e) p1 *= A_DEC; }
        Av0[k4] = p0; Av1[k4] = p1;
    }

    // ---- per-lane carry powers a^(rowD+1), rowD = 8*half + r ---------------
    float apow[8];
    {
        float p = A_DEC;
        for (int e = 0; e < 8 * half; ++e) p *= A_DEC;
#pragma unroll
        for (int r = 0; r < 8; ++r) { apow[r] = p; p *= A_DEC; }
    }

    for (int blk = 0; blk < NB; ++blk) {
        float* cur = lds[wv][blk & 1];

        if (blk + 1 < NB) {
            // Prefetch next block into the other buffer (overlaps compute).
            const unsigned la0 = ((blk + 1) & 1) ? laB0 : laA0;
            const unsigned la1 = ((blk + 1) & 1) ? laB1 : laA1;
            asm volatile("s_wait_dscnt 0" ::: "memory");  // prior reads done
            asm volatile("global_load_async_to_lds_b128 %0, %1, off"
                         :: "v"(la0), "v"((unsigned long long)(size_t)gq0)
                         : "memory");
            asm volatile("global_load_async_to_lds_b128 %0, %1, off"
                         :: "v"(la1), "v"((unsigned long long)(size_t)gq1)
                         : "memory");
            gq0 += (size_t)16 * N_;
            gq1 += (size_t)16 * N_;
            // In-order completion: <=2 outstanding => current block resident.
            asm volatile("s_wait_asynccnt 2" ::: "memory");
        } else {
            asm volatile("s_wait_asynccnt 0" ::: "memory");
        }

        // Drive matrix slices + 16x16x16 matmul on the WMMA unit.
        v8f acc = {0.f, 0.f, 0.f, 0.f, 0.f, 0.f, 0.f, 0.f};
#pragma unroll
        for (int k4 = 0; k4 < 4; ++k4) {
            const int r0 = 4 * k4 + 2 * half;
            float i0;
            if (k4 == 0) { const float tl = cur[16 + m]; i0 = half ? tl : prevm; }
            else          i0 = cur[(r0 - 1) * 16 + m];
            const float i1 = cur[r0 * 16 + m];
            const float i2 = cur[(r0 + 1) * 16 + m];
            v2f Bv; Bv.x = KD * fabsf(i1 - i0);           // u[tb + r0]
                    Bv.y = KD * fabsf(i2 - i1);           // u[tb + r0 + 1]
            v2f Aq; Aq.x = Av0[k4]; Aq.y = Av1[k4];
            acc = __builtin_amdgcn_wmma_f32_16x16x4_f32(
                      false, Aq, false, Bv, (short)0, acc, false, false);
        }
        prevm = cur[15 * 16 + m];                         // boundary for next blk

        // Carry term + non-temporal stores: single base + r*8192B immediate
        // offsets so the 8 stores form one clause.
        float y[8];
#pragma unroll
        for (int r = 0; r < 8; ++r) {
            y[r] = fmaf(s, apow[r], acc[r]);
            __builtin_nontemporal_store(y[r], op + (size_t)r * N_);
        }
        op += (size_t)16 * N_;
        // New carry = y at rowD==15 (hi half, r==7); broadcast to both halves.
        s = __shfl(y[7], 16 + m, 32);
    }
}

// ---------------------------------------------------------------------------
extern "C" void kernel_launch(void* const* d_in, const int* in_sizes, int n_in,
                              void* d_out, int out_size, void* d_ws, size_t ws_size,
                              hipStream_t stream) {
    (void)in_sizes; (void)n_in; (void)out_size; (void)ws_size;
    const float* in  = (const float*)d_in[0];
    float*       out = (float*)d_out;
    float*       fin = (float*)d_ws;       // CCH * BN floats = 1 MiB scratch

    dim3 gA(BN / 4 / 256, CCH);
    ra_finals_kernel<<<gA, 256, 0, stream>>>(in, fin);

    const int totalTiles = (BN / 16) * CCH;               // 16384 wave-tiles
    ra_scan_kernel<<<totalTiles / WPB, 32 * WPB, 0, stream>>>(in, fin, out);
}